// GATA_67199058313304
// MI455X (gfx1250) — compile-verified
//
#include <hip/hip_runtime.h>
#include <hip/hip_bf16.h>
#include <math.h>

// Problem constants (from reference)
#define BQ   2
#define NQ   512
#define JQ   32
#define DQ   256
#define HQ   8
#define MQ   5
#define MDQ  1280   // M*D
#define PI_F 3.14159265358979f

typedef __attribute__((ext_vector_type(2))) float v2f;
typedef __attribute__((ext_vector_type(8))) float v8f;

__device__ __forceinline__ float silu_f(float v) { return v / (1.f + expf(-v)); }

#define V8F_ZERO {0.f,0.f,0.f,0.f,0.f,0.f,0.f,0.f}

// 1x4 register-blocked WMMA: one 16-row tile x four 16-col tiles, K=256.
// A fragment (float2/lane) is loaded once per K-step and reused by 4 WMMAs,
// giving 1.25 b64 loads per v_wmma and 4 independent accumulation chains.
// Fragment layout (f32 16x16x4): lane L, vgpr v <-> row = L%16, K = 2*(L>>4)+v.
__device__ __forceinline__ void wmma_row4_k256(const float* __restrict__ arow,
                                               const float* __restrict__ w0,
                                               const float* __restrict__ w1,
                                               const float* __restrict__ w2,
                                               const float* __restrict__ w3,
                                               v8f acc[4]) {
  #pragma unroll 2
  for (int k0 = 0; k0 < DQ; k0 += 4) {
    v2f a  = *(const v2f*)(arow + k0);
    v2f b0 = *(const v2f*)(w0 + k0);
    v2f b1 = *(const v2f*)(w1 + k0);
    v2f b2 = *(const v2f*)(w2 + k0);
    v2f b3 = *(const v2f*)(w3 + k0);
    acc[0] = __builtin_amdgcn_wmma_f32_16x16x4_f32(false, a, false, b0, (short)0, acc[0], false, false);
    acc[1] = __builtin_amdgcn_wmma_f32_16x16x4_f32(false, a, false, b1, (short)0, acc[1], false, false);
    acc[2] = __builtin_amdgcn_wmma_f32_16x16x4_f32(false, a, false, b2, (short)0, acc[2], false, false);
    acc[3] = __builtin_amdgcn_wmma_f32_16x16x4_f32(false, a, false, b3, (short)0, acc[3], false, false);
  }
}

// ---------------- LayerNorm over D=256 ----------------
__global__ __launch_bounds__(256) void ln_kernel(const float* __restrict__ h,
                                                 const float* __restrict__ w,
                                                 const float* __restrict__ b,
                                                 float* __restrict__ hn) {
  __shared__ float red[256];
  int row = blockIdx.x, d = threadIdx.x;
  float v = h[row * DQ + d];
  red[d] = v; __syncthreads();
  for (int s = 128; s > 0; s >>= 1) { if (d < s) red[d] += red[d + s]; __syncthreads(); }
  float mu = red[0] * (1.f / DQ);
  __syncthreads();
  float c = v - mu;
  red[d] = c * c; __syncthreads();
  for (int s = 128; s > 0; s >>= 1) { if (d < s) red[d] += red[d + s]; __syncthreads(); }
  float var = red[0] * (1.f / DQ);
  hn[row * DQ + d] = c * rsqrtf(var + 1e-5f) * w[d] + b[d];
}

// ---- Generic GEMM: out[R,C] = act(A[R,256] @ W[C,256]^T), wave = 16 rows x 64 cols ----
__global__ __launch_bounds__(256) void gemm_wmma(const float* __restrict__ A,
                                                 const float* __restrict__ W,
                                                 float* __restrict__ out,
                                                 int R, int C, int act) {
  int wg   = blockIdx.x * 8 + (threadIdx.x >> 5);   // wave-group: (rt, 4-col group)
  int lane = threadIdx.x & 31;
  int gC   = C >> 6;                                // groups of 4 col-tiles
  int total = (R >> 4) * gC;
  if (wg >= total) return;                          // wave-uniform: EXEC all-1s for WMMA
  int rt = wg / gC, ct0 = (wg % gC) * 4;
  int lo = lane & 15, hi = lane >> 4;
  const float* arow = A + (size_t)(rt * 16 + lo) * DQ + hi * 2;
  const float* wbase = W + (size_t)(ct0 * 16 + lo) * DQ + hi * 2;
  v8f acc[4] = {V8F_ZERO, V8F_ZERO, V8F_ZERO, V8F_ZERO};
  wmma_row4_k256(arow, wbase, wbase + 16 * DQ, wbase + 32 * DQ, wbase + 48 * DQ, acc);
  #pragma unroll
  for (int i = 0; i < 4; ++i) {
    float* obase = out + (size_t)(rt * 16 + 8 * hi) * C + (ct0 + i) * 16 + lo;
    #pragma unroll
    for (int r = 0; r < 8; ++r) {                   // row = rt*16 + r + 8*hi
      float v = acc[i][r];
      if (act) v = silu_f(v);
      obase[(size_t)r * C] = v;
    }
  }
}

// -------- Fused t_attn (WMMA+silu into LDS) + attention logits + softmax --------
// One block per (b,i). Wave w: rt = w&1, ct0 = (w>>1)*4  (covers all 2x16 tiles).
__global__ __launch_bounds__(256) void attn_kernel(const float* __restrict__ t_ij,
                                                   const float* __restrict__ W_re,
                                                   const float* __restrict__ q,
                                                   const float* __restrict__ k,
                                                   const int* __restrict__ nidx,
                                                   const int* __restrict__ nmask,
                                                   float* __restrict__ attn) {
  __shared__ float ta[JQ][DQ + 1];     // +1 pad: conflict-free row access
  __shared__ int   nbrs[JQ];
  int bi = blockIdx.x;
  int tid = threadIdx.x, lane = tid & 31, wave = tid >> 5;
  if (tid < JQ) nbrs[tid] = nidx[bi * JQ + tid];
  int lo = lane & 15, hi = lane >> 4;
  int rt = wave & 1, ct0 = (wave >> 1) * 4;
  const float* arow = t_ij + (size_t)bi * JQ * DQ + (size_t)(rt * 16 + lo) * DQ + hi * 2;
  const float* wbase = W_re + (size_t)(ct0 * 16 + lo) * DQ + hi * 2;
  v8f acc[4] = {V8F_ZERO, V8F_ZERO, V8F_ZERO, V8F_ZERO};
  wmma_row4_k256(arow, wbase, wbase + 16 * DQ, wbase + 32 * DQ, wbase + 48 * DQ, acc);
  #pragma unroll
  for (int i = 0; i < 4; ++i)
    #pragma unroll
    for (int r = 0; r < 8; ++r)
      ta[rt * 16 + r + 8 * hi][(ct0 + i) * 16 + lo] = silu_f(acc[i][r]);
  __syncthreads();

  int h = wave, j = lane;
  int nb = nbrs[j];
  int b = bi >> 9;                              // bi / N
  const float* qrow = q + (size_t)bi * DQ + h * 32;
  const float* krow = k + (size_t)(b * NQ + nb) * DQ + h * 32;
  float logit = 0.f;
  #pragma unroll 8
  for (int dd = 0; dd < 32; ++dd)
    logit += qrow[dd] * krow[dd] * ta[j][h * 32 + dd];
  int msk = nmask[bi * JQ + j];
  logit = msk ? logit : -1e9f;
  // wave32 softmax over j (matches jax max-subtracted softmax)
  float mx = logit;
  for (int off = 16; off; off >>= 1) mx = fmaxf(mx, __shfl_xor(mx, off, 32));
  float e = expf(logit - mx);
  float sum = e;
  for (int off = 16; off; off >>= 1) sum += __shfl_xor(sum, off, 32);
  unsigned long long bal = __ballot(msk != 0);
  float norm = sqrtf((float)__popcll(bal)) * (1.f / 16.f);   // sqrt(n)/sqrt(256)
  attn[(size_t)bi * HQ * JQ + h * JQ + j] = (e / sum) * norm;
}

// -------- Fused t_filt WMMA + spatial/sea elementwise + j-reduction --------
__global__ __launch_bounds__(256) void fused_edge_kernel(
    const float* __restrict__ t_ij, const float* __restrict__ W_rs,
    const float* __restrict__ xg,   const float* __restrict__ vg,
    const float* __restrict__ attn,
    const float* __restrict__ r1,   const float* __restrict__ r2,
    const float* __restrict__ r_ij, const int* __restrict__ nidx,
    const float* __restrict__ X1,   const float* __restrict__ X2,
    float* __restrict__ dH, float* __restrict__ dX1, float* __restrict__ dX2) {
  __shared__ float tf[JQ][DQ + 1];
  __shared__ float aL[HQ][JQ];
  __shared__ float cutL[JQ];
  __shared__ float r1L[JQ][3];
  __shared__ float r2L[JQ][5];
  __shared__ int   nbrs[JQ];
  int bi = blockIdx.x, b = bi >> 9;
  int tid = threadIdx.x, lane = tid & 31, wave = tid >> 5;
  if (tid < JQ) {
    nbrs[tid] = nidx[bi * JQ + tid];
    float r = r_ij[bi * JQ + tid];
    cutL[tid] = (r < 5.0f) ? 0.5f * (cosf(r * (PI_F / 5.0f)) + 1.f) : 0.f;
  }
  if (tid < JQ * 3) r1L[tid / 3][tid % 3] = r1[(size_t)bi * JQ * 3 + tid];
  if (tid < JQ * 5) r2L[tid / 5][tid % 5] = r2[(size_t)bi * JQ * 5 + tid];
  aL[wave][lane] = attn[(size_t)bi * HQ * JQ + tid];
  __syncthreads();

  int d = tid;
  float accH = 0.f;
  float aX1[3] = {0.f, 0.f, 0.f};
  float aX2[5] = {0.f, 0.f, 0.f, 0.f, 0.f};
  int lo = lane & 15, hi = lane >> 4;
  int rt = wave & 1, ct0 = (wave >> 1) * 4;
  const float* arow = t_ij + (size_t)bi * JQ * DQ + (size_t)(rt * 16 + lo) * DQ + hi * 2;
  const float* wtile = W_rs + (size_t)(ct0 * 16 + lo) * DQ + hi * 2;   // advances by m*DQ rows

  for (int m = 0; m < MQ; ++m) {
    // ---- WMMA: tf[0..31][0..255] = t_ij(b,i) @ W_rs[m*256 : (m+1)*256]^T ----
    const float* wb = wtile + (size_t)m * DQ * DQ;
    if (m + 1 < MQ) {                       // warm GL2 for the next m-plane of W_rs
      __builtin_prefetch(wb + (size_t)DQ * DQ, 0, 1);
      __builtin_prefetch(wb + (size_t)DQ * DQ + 16 * DQ, 0, 1);
      __builtin_prefetch(wb + (size_t)DQ * DQ + 32 * DQ, 0, 1);
      __builtin_prefetch(wb + (size_t)DQ * DQ + 48 * DQ, 0, 1);
    }
    v8f acc[4] = {V8F_ZERO, V8F_ZERO, V8F_ZERO, V8F_ZERO};
    wmma_row4_k256(arow, wb, wb + 16 * DQ, wb + 32 * DQ, wb + 48 * DQ, acc);
    #pragma unroll
    for (int i = 0; i < 4; ++i)
      #pragma unroll
      for (int r = 0; r < 8; ++r)
        tf[rt * 16 + r + 8 * hi][(ct0 + i) * 16 + lo] = acc[i][r];
    __syncthreads();

    // ---- consume: out[j][e] = tf*x_j*cut + attn[h(e)]*v_j,  e = m*256+d ----
    int e = m * DQ + d;
    int h = e / 160;                 // sea layout: v reshaped as (H=8, M*dh=160)
    if (m == 0) {
      float s = 0.f;
      for (int j = 0; j < JQ; ++j) {
        size_t gi = (size_t)(b * NQ + nbrs[j]) * MDQ + e;
        s += tf[j][d] * xg[gi] * cutL[j] + aL[h][j] * vg[gi];
      }
      accH = s;
    } else if (m == 1) {
      for (int j = 0; j < JQ; ++j) {
        size_t gi = (size_t)(b * NQ + nbrs[j]) * MDQ + e;
        float o = tf[j][d] * xg[gi] * cutL[j] + aL[h][j] * vg[gi];
        aX1[0] += r1L[j][0] * o; aX1[1] += r1L[j][1] * o; aX1[2] += r1L[j][2] * o;
      }
    } else if (m == 2) {
      for (int j = 0; j < JQ; ++j) {
        size_t gi = (size_t)(b * NQ + nbrs[j]) * MDQ + e;
        float o = tf[j][d] * xg[gi] * cutL[j] + aL[h][j] * vg[gi];
        #pragma unroll
        for (int c = 0; c < 5; ++c) aX2[c] += r2L[j][c] * o;
      }
    } else if (m == 3) {
      for (int j = 0; j < JQ; ++j) {
        int nb = nbrs[j];
        size_t gi = (size_t)(b * NQ + nb) * MDQ + e;
        float o = tf[j][d] * xg[gi] * cutL[j] + aL[h][j] * vg[gi];
        size_t x1b = (size_t)((b * NQ + nb) * 3) * DQ + d;
        aX1[0] += X1[x1b] * o;
        aX1[1] += X1[x1b + DQ] * o;
        aX1[2] += X1[x1b + 2 * DQ] * o;
      }
    } else {
      for (int j = 0; j < JQ; ++j) {
        int nb = nbrs[j];
        size_t gi = (size_t)(b * NQ + nb) * MDQ + e;
        float o = tf[j][d] * xg[gi] * cutL[j] + aL[h][j] * vg[gi];
        size_t x2b = (size_t)((b * NQ + nb) * 5) * DQ + d;
        #pragma unroll
        for (int c = 0; c < 5; ++c) aX2[c] += X2[x2b + (size_t)c * DQ] * o;
      }
    }
    __syncthreads();   // before next m overwrites tf
  }

  dH[(size_t)bi * DQ + d] = accH;
  #pragma unroll
  for (int c = 0; c < 3; ++c) dX1[((size_t)bi * 3 + c) * DQ + d] = aX1[c];
  #pragma unroll
  for (int c = 0; c < 5; ++c) dX2[((size_t)bi * 5 + c) * DQ + d] = aX2[c];
}

extern "C" void kernel_launch(void* const* d_in, const int* in_sizes, int n_in,
                              void* d_out, int out_size, void* d_ws, size_t ws_size,
                              hipStream_t stream) {
  const float* h    = (const float*)d_in[0];
  const float* X1   = (const float*)d_in[1];
  const float* X2   = (const float*)d_in[2];
  const float* r1   = (const float*)d_in[3];
  const float* r2   = (const float*)d_in[4];
  const float* tij  = (const float*)d_in[5];
  const float* rij  = (const float*)d_in[6];
  const int*   nidx = (const int*)d_in[7];
  const int*   nmsk = (const int*)d_in[8];
  const float* lnw  = (const float*)d_in[9];
  const float* lnb  = (const float*)d_in[10];
  const float* Wq   = (const float*)d_in[11];
  const float* Wk   = (const float*)d_in[12];
  const float* Wre  = (const float*)d_in[13];
  const float* Wrs  = (const float*)d_in[14];
  const float* gs1  = (const float*)d_in[15];
  const float* gs2  = (const float*)d_in[16];
  const float* gv1  = (const float*)d_in[17];
  const float* gv2  = (const float*)d_in[18];

  // workspace layout (floats); total 4,194,304 floats = 16 MiB
  float* ws = (float*)d_ws;
  float* hn = ws;                 // B*N*D      = 262144
  float* q  = ws + 262144;        // 262144
  float* k  = ws + 524288;        // 262144
  float* s1 = ws + 786432;        // 262144
  float* v1 = ws + 1048576;       // 262144
  float* xb = ws + 1310720;       // B*N*MD     = 1310720
  float* vb = ws + 2621440;       // 1310720
  float* at = ws + 3932160;       // B*N*H*J    = 262144

  float* dH  = (float*)d_out;     // 262144
  float* dX1 = dH + 262144;       // 786432
  float* dX2 = dX1 + 786432;      // 1310720

  dim3 blk(256);
  ln_kernel<<<BQ * NQ, blk, 0, stream>>>(h, lnw, lnb, hn);
  // node GEMMs: R=1024 rows, K=256. wave-group = 16x64 tile:
  //  C=256:  64 rt * 4 groups  = 256 wave-groups -> 32 blocks
  //  C=1280: 64 rt * 20 groups = 1280 wave-groups -> 160 blocks
  gemm_wmma<<<32,  blk, 0, stream>>>(hn, Wq,  q,  BQ * NQ, DQ,  0);
  gemm_wmma<<<32,  blk, 0, stream>>>(hn, Wk,  k,  BQ * NQ, DQ,  0);
  gemm_wmma<<<32,  blk, 0, stream>>>(hn, gs1, s1, BQ * NQ, DQ,  1);
  gemm_wmma<<<32,  blk, 0, stream>>>(hn, gv1, v1, BQ * NQ, DQ,  1);
  gemm_wmma<<<160, blk, 0, stream>>>(s1, gs2, xb, BQ * NQ, MDQ, 0);
  gemm_wmma<<<160, blk, 0, stream>>>(v1, gv2, vb, BQ * NQ, MDQ, 0);
  // fused t_attn + logits + softmax
  attn_kernel<<<BQ * NQ, blk, 0, stream>>>(tij, Wre, q, k, nidx, nmsk, at);
  // fused t_filt + spatial/sea + reductions
  fused_edge_kernel<<<BQ * NQ, blk, 0, stream>>>(tij, Wrs, xb, vb, at, r1, r2,
                                                 rij, nidx, X1, X2, dH, dX1, dX2);
}